// obj_feature_fuser_66700842107135
// MI455X (gfx1250) — compile-verified
//
#include <hip/hip_runtime.h>
#include <hip/hip_bf16.h>
#include <math.h>
#include <stdint.h>

#define NB    16
#define NRr   300
#define NRSO  300
#define CC    150
#define NREL  50
#define DPRO  512
#define DROI  12544
#define DREL  256
#define KK    10
#define RRk   3
#define R2k   9
#define MAXR  100
#define NTRI  (NRSO * R2k)   // 2700
#define OUTROWS 320          // 300 + 10 + 10

typedef unsigned int v4u_ __attribute__((ext_vector_type(4)));
typedef int          v8i_ __attribute__((ext_vector_type(8)));
typedef int          v4i_ __attribute__((ext_vector_type(4)));

__device__ __forceinline__ float sigmoidf_(float x) { return 1.0f / (1.0f + expf(-x)); }

// ---------------------------------------------------------------------------
// Kernel 1: all the selection logic. One block per batch element.
// ---------------------------------------------------------------------------
__global__ __launch_bounds__(1024)
void compute_select(const float* __restrict__ so_cls,   // (16,300,300)
                    const float* __restrict__ pbso,     // (16,300,8)
                    const float* __restrict__ rel_log,  // (16,300,50)
                    const float* __restrict__ cls_log,  // (16,600,150)
                    const float* __restrict__ pbx,      // (16,600,4)
                    const unsigned char* __restrict__ vis, // (16,300) bytes (zeros)
                    int* __restrict__ ent_out,          // (16,10) -> workspace
                    float* __restrict__ vis_out)        // (16,300) floats
{
    const int n = blockIdx.x;
    const int tid = threadIdx.x;
    const int bd = blockDim.x;

    __shared__ float s_ss[NRSO][RRk]; __shared__ int s_sc[NRSO][RRk];
    __shared__ float s_os[NRSO][RRk]; __shared__ int s_oc[NRSO][RRk];
    __shared__ float s_rm[NRSO];
    __shared__ float s_tri[NTRI];
    __shared__ unsigned char s_flag[NTRI];
    __shared__ int   s_selc[2 * MAXR];
    __shared__ float s_selb[2 * MAXR][4];
    __shared__ float s_pm[NRr]; __shared__ int s_pc[NRr];
    __shared__ float s_score[NRr];
    __shared__ unsigned char s_esel[NRr];

    // Phase 1: per relation pair — top-3 subject / top-3 object class probs, rel_max.
    for (int i = tid; i < NRSO; i += bd) {
        const float* base = so_cls + ((size_t)n * NRSO + i) * (2 * CC);
        float v0 = -1e30f, v1 = -1e30f, v2 = -1e30f; int c0 = 0, c1 = 0, c2 = 0;
        for (int c = 0; c < CC; c++) {
            float v = sigmoidf_(base[c]);
            if (v > v0)      { v2 = v1; c2 = c1; v1 = v0; c1 = c0; v0 = v; c0 = c; }
            else if (v > v1) { v2 = v1; c2 = c1; v1 = v;  c1 = c; }
            else if (v > v2) { v2 = v;  c2 = c; }
        }
        s_ss[i][0] = v0; s_ss[i][1] = v1; s_ss[i][2] = v2;
        s_sc[i][0] = c0; s_sc[i][1] = c1; s_sc[i][2] = c2;
        v0 = v1 = v2 = -1e30f; c0 = c1 = c2 = 0;
        for (int c = 0; c < CC; c++) {
            float v = sigmoidf_(base[CC + c]);
            if (v > v0)      { v2 = v1; c2 = c1; v1 = v0; c1 = c0; v0 = v; c0 = c; }
            else if (v > v1) { v2 = v1; c2 = c1; v1 = v;  c1 = c; }
            else if (v > v2) { v2 = v;  c2 = c; }
        }
        s_os[i][0] = v0; s_os[i][1] = v1; s_os[i][2] = v2;
        s_oc[i][0] = c0; s_oc[i][1] = c1; s_oc[i][2] = c2;
        const float* rb = rel_log + ((size_t)n * NRSO + i) * NREL;
        float m = rb[0];
        for (int r = 1; r < NREL; r++) m = fmaxf(m, rb[r]);
        s_rm[i] = sigmoidf_(m);   // sigmoid monotone: max(sigmoid(x)) == sigmoid(max x)
    }
    __syncthreads();

    // Phase 2: triple scores, 2700 combos (X = c/3 subject rank, Y = c%3 object rank).
    for (int idx = tid; idx < NTRI; idx += bd) {
        int t = idx / R2k, c = idx % R2k;
        s_tri[idx] = s_rm[t] * s_ss[t][c / RRk] * s_os[t][c % RRk];
    }
    __syncthreads();

    // Phase 3: exact top-100 via rank counting (jax tie-break: lower index wins).
    for (int idx = tid; idx < NTRI; idx += bd) {
        float v = s_tri[idx]; int rank = 0;
        for (int j = 0; j < NTRI; j++) {
            float vj = s_tri[j];
            rank += (vj > v) || (vj == v && j < idx);
        }
        s_flag[idx] = (rank < MAXR) ? 1 : 0;
    }
    __syncthreads();
    for (int idx = tid; idx < NTRI; idx += bd) if (s_flag[idx]) {
        int pos = 0;
        for (int j = 0; j < idx; j++) pos += s_flag[j];   // index-ascending == jnp.sort(tri_idx)
        int t = idx / R2k, c = idx % R2k;
        s_selc[2 * pos]     = s_sc[t][c / RRk];
        s_selc[2 * pos + 1] = s_oc[t][c % RRk];
        const float* bb = pbso + ((size_t)n * NRSO + t) * 8;
        s_selb[2 * pos][0] = bb[0]; s_selb[2 * pos][1] = bb[1];
        s_selb[2 * pos][2] = bb[2]; s_selb[2 * pos][3] = bb[3];
        s_selb[2 * pos + 1][0] = bb[4]; s_selb[2 * pos + 1][1] = bb[5];
        s_selb[2 * pos + 1][2] = bb[6]; s_selb[2 * pos + 1][3] = bb[7];
    }
    __syncthreads();

    // Phase 4: per proposal (subject rows of class_logits): max/argmax of sigmoid.
    for (int j = tid; j < NRr; j += bd) {
        const float* base = cls_log + ((size_t)n * 2 * NRr + 2 * j) * CC;
        float best = -1e30f; int bi = 0;
        for (int c = 0; c < CC; c++) {
            float v = sigmoidf_(base[c]);
            if (v > best) { best = v; bi = c; }
        }
        s_pm[j] = best; s_pc[j] = bi;
    }
    __syncthreads();

    // Phase 5: is_new via IoU>=0.5 & class equality against the 200 selected boxes.
    for (int j = tid; j < NRr; j += bd) {
        const float* b1 = pbx + ((size_t)n * 2 * NRr + 2 * j) * 4;
        float x1 = b1[0], y1 = b1[1], x2 = b1[2], y2 = b1[3];
        float a1 = (y2 - y1 + 1.0f) * (x2 - x1 + 1.0f);
        int pc = s_pc[j];
        bool hit = false;
        for (int k = 0; k < 2 * MAXR; k++) {
            if (s_selc[k] != pc) continue;
            float bx1 = s_selb[k][0], by1 = s_selb[k][1];
            float bx2 = s_selb[k][2], by2 = s_selb[k][3];
            float a2 = (by2 - by1 + 1.0f) * (bx2 - bx1 + 1.0f);
            float w = fmaxf(fminf(x2, bx2) - fmaxf(x1, bx1), 0.0f);
            float h = fmaxf(fminf(y2, by2) - fmaxf(y1, by1), 0.0f);
            float inter = w * h;
            float iou = inter / (a1 + a2 - inter);
            if (iou >= 0.5f) { hit = true; break; }
        }
        float sc = s_pm[j];
        if (hit) sc -= 2.0f;
        if (vis[(size_t)n * NRr + j]) sc -= 2.0f;
        s_score[j] = sc;
    }
    __syncthreads();

    // Phase 6: exact top-10 entity selection, sorted ascending; vis_out.
    for (int j = tid; j < NRr; j += bd) {
        float v = s_score[j]; int rank = 0;
        for (int i = 0; i < NRr; i++) {
            float vi = s_score[i];
            rank += (vi > v) || (vi == v && i < j);
        }
        s_esel[j] = (rank < KK) ? 1 : 0;
    }
    __syncthreads();
    for (int j = tid; j < NRr; j += bd) {
        bool sel = s_esel[j] != 0;
        if (sel) {
            int pos = 0;
            for (int i = 0; i < j; i++) pos += s_esel[i];
            ent_out[n * KK + pos] = j;
        }
        bool v = vis[(size_t)n * NRr + j] != 0;
        vis_out[(size_t)n * NRr + j] = (v || sel) ? 1.0f : 0.0f;
    }
}

// ---------------------------------------------------------------------------
// Kernel 2: generic B128 batched 2-D copy (src: 300 rows/batch -> dst: rows
// 0..299 of 320-row blocks). rowV4 = row length in float4 units.
// ---------------------------------------------------------------------------
__global__ void copy2d_v4(float4* __restrict__ dst, const float4* __restrict__ src, int rowV4)
{
    const long per = (long)NRr * rowV4;
    const long total = (long)NB * per;
    const long stride = (long)gridDim.x * blockDim.x;
    for (long idx = (long)blockIdx.x * blockDim.x + threadIdx.x; idx < total; idx += stride) {
        long nb = idx / per;
        long rem = idx - nb * per;
        dst[nb * (long)OUTROWS * rowV4 + rem] = src[idx];
    }
}

// ---------------------------------------------------------------------------
// Kernel 3: bulk ROI copy via the Tensor Data Mover (CDNA5 TDM).
// One wave per 50 KB row: TENSOR_LOAD_TO_LDS -> s_wait_tensorcnt ->
// TENSOR_STORE_FROM_LDS. Falls back to B128 VALU copy if the builtin is
// missing on the compiling toolchain.
// ---------------------------------------------------------------------------
__global__ __launch_bounds__(32)
void tdm_copy_roi(const float* __restrict__ src, float* __restrict__ dst)
{
    const int b = blockIdx.x;
    const int n = b / NRr, r = b - n * NRr;
    const float* s = src + ((size_t)n * NRr + r) * DROI;
    float*       d = dst + ((size_t)n * OUTROWS + r) * DROI;
#if defined(__AMDGCN__) && __has_builtin(__builtin_amdgcn_tensor_load_to_lds) && \
    __has_builtin(__builtin_amdgcn_tensor_store_from_lds) && \
    __has_builtin(__builtin_amdgcn_s_wait_tensorcnt)
    __shared__ __align__(16) float buf[DROI];       // 50176 B staging tile
    uint64_t sA = (uint64_t)(uintptr_t)s;
    uint64_t dA = (uint64_t)(uintptr_t)d;
    uint32_t lds = (uint32_t)(uintptr_t)&buf[0];    // low 32 bits of shared-aperture addr = LDS offset

    // D# group1 (256b): data_size=4B, tensor_dim0=tile_dim0=12544, tensor_dim1=tile_dim1=1,
    // tensor_dim0_stride=12544. 1-D copy, no padding/iterate/multicast.
    v8i_ g1;
    g1[0] = 0x00020000;                              // data_size = 2 (4 bytes)
    g1[1] = (int)(((unsigned)DROI & 0xFFFFu) << 16); // tensor_dim0[15:0]
    g1[2] = (int)(1u << 16);                         // tensor_dim0[31:16]=0 | tensor_dim1[15:0]=1
    g1[3] = (int)(((unsigned)DROI & 0xFFFFu) << 16); // tensor_dim1[31:16]=0 | tile_dim0
    g1[4] = 1;                                       // tile_dim1=1, tile_dim2=0
    g1[5] = DROI;                                    // tensor_dim0_stride[31:0]
    g1[6] = 0;
    g1[7] = 0;
    v4i_ z4 = {0, 0, 0, 0};                          // groups 2/3 unused (2-D descriptor)
    v8i_ z8 = {0, 0, 0, 0, 0, 0, 0, 0};              // extra group (clang-23 6-arg form)

    v4u_ g0;                                         // group0: count=1, lds, 57b global addr, type=2
    g0[0] = 1u;
    g0[1] = lds;
    g0[2] = (uint32_t)sA;
    g0[3] = ((uint32_t)(sA >> 32) & 0x01FFFFFFu) | (2u << 30);
    __builtin_amdgcn_tensor_load_to_lds(g0, g1, z4, z4, z8, 0);
    __builtin_amdgcn_s_wait_tensorcnt(0);

    g0[2] = (uint32_t)dA;
    g0[3] = ((uint32_t)(dA >> 32) & 0x01FFFFFFu) | (2u << 30);
    __builtin_amdgcn_tensor_store_from_lds(g0, g1, z4, z4, z8, 0);
    __builtin_amdgcn_s_wait_tensorcnt(0);
#else
    const float4* s4 = (const float4*)s;
    float4*       d4 = (float4*)d;
    for (int c = threadIdx.x; c < DROI / 4; c += 32) d4[c] = s4[c];
#endif
}

// ---------------------------------------------------------------------------
// Kernel 4: gathered aux rows (300..319) for all five feature outputs,
// including the half-dim swaps. Tiny (~17 MB), scalar is fine.
// Payload per (batch, slot): 2*(512 + 8 + 300 + 256 + 12544) = 27240 floats.
// ---------------------------------------------------------------------------
__global__ void gather_aux(const float* __restrict__ pro,      // (16,300,512)
                           const float* __restrict__ pbx,      // (16,600,4)
                           const float* __restrict__ cls_log,  // (16,600,150)
                           const float* __restrict__ auxrel,   // (16,300,512)
                           const float* __restrict__ roi,      // (16,300,12544)
                           const int* __restrict__ ent,        // (16,10)
                           float* __restrict__ out0, float* __restrict__ out1,
                           float* __restrict__ out2, float* __restrict__ out3,
                           float* __restrict__ out4)
{
    const long P = 2L * (DPRO + 8 + 2 * CC + DREL + DROI);  // 27240
    const long total = (long)NB * KK * P;
    const long stride = (long)gridDim.x * blockDim.x;
    for (long idx = (long)blockIdx.x * blockDim.x + threadIdx.x; idx < total; idx += stride) {
        long ns = idx / P;
        int e = (int)(idx - ns * P);
        int n = (int)(ns / KK);
        int s = (int)(ns - (long)n * KK);
        int j = ent[n * KK + s];
        if (e < 2 * DPRO) {                                  // so_pro aux + swap
            const float* srow = pro + ((size_t)n * NRr + j) * DPRO;
            int half = e >= DPRO; int c = e - half * DPRO;
            int sc = half ? (c + DPRO / 2) % DPRO : c;
            out0[((size_t)n * OUTROWS + 300 + half * 10 + s) * DPRO + c] = srow[sc];
        } else if ((e -= 2 * DPRO) < 16) {                   // boxes aux + swap
            const float* srow = pbx + ((size_t)n * NRr + j) * 8;
            int half = e >= 8; int c = e - half * 8;
            int sc = half ? (c + 4) % 8 : c;
            out1[((size_t)n * OUTROWS + 300 + half * 10 + s) * 8 + c] = srow[sc];
        } else if ((e -= 16) < 4 * CC) {                     // cls aux + swap
            const float* srow = cls_log + ((size_t)n * NRr + j) * (2 * CC);
            int half = e >= 2 * CC; int c = e - half * 2 * CC;
            int sc = half ? (c + CC) % (2 * CC) : c;
            out2[((size_t)n * OUTROWS + 300 + half * 10 + s) * (2 * CC) + c] = srow[sc];
        } else if ((e -= 4 * CC) < 2 * DREL) {               // rel aux halves (no swap)
            const float* srow = auxrel + ((size_t)n * NRr + j) * (2 * DREL);
            int half = e >= DREL; int c = e - half * DREL;
            out3[((size_t)n * OUTROWS + 300 + half * 10 + s) * DREL + c] = srow[half * DREL + c];
        } else {                                             // roi aux + swap
            e -= 2 * DREL;
            const float* srow = roi + ((size_t)n * NRr + j) * DROI;
            int half = e >= DROI; int c = e - half * DROI;
            int sc = half ? (c + DROI / 2) % DROI : c;
            out4[((size_t)n * OUTROWS + 300 + half * 10 + s) * DROI + c] = srow[sc];
        }
    }
}

// ---------------------------------------------------------------------------
extern "C" void kernel_launch(void* const* d_in, const int* in_sizes, int n_in,
                              void* d_out, int out_size, void* d_ws, size_t ws_size,
                              hipStream_t stream)
{
    const float* auxrel   = (const float*)d_in[0];
    const float* cls_log  = (const float*)d_in[1];
    const float* pbx      = (const float*)d_in[2];
    const float* pro      = (const float*)d_in[3];
    const float* roi      = (const float*)d_in[4];
    const float* so_cls   = (const float*)d_in[5];
    const float* pbso     = (const float*)d_in[6];
    const float* so_pro   = (const float*)d_in[7];
    const float* so_roi   = (const float*)d_in[8];
    const float* rel_log  = (const float*)d_in[9];
    const float* rel_feat = (const float*)d_in[10];
    const unsigned char* vis = (const unsigned char*)d_in[11];

    float* out  = (float*)d_out;
    float* out0 = out;                                   // (16,320,512)
    float* out1 = out0 + (size_t)NB * OUTROWS * DPRO;    // (16,320,8)
    float* out2 = out1 + (size_t)NB * OUTROWS * 8;       // (16,320,300)
    float* out3 = out2 + (size_t)NB * OUTROWS * 2 * CC;  // (16,320,256)
    float* out4 = out3 + (size_t)NB * OUTROWS * DREL;    // (16,320,12544)
    float* out5 = out4 + (size_t)NB * OUTROWS * DROI;    // (16,300) vis

    int* ent = (int*)d_ws;

    compute_select<<<NB, 1024, 0, stream>>>(so_cls, pbso, rel_log, cls_log, pbx, vis, ent, out5);

    copy2d_v4<<<2400, 256, 0, stream>>>((float4*)out0, (const float4*)so_pro,   DPRO / 4);
    copy2d_v4<<<40,   256, 0, stream>>>((float4*)out1, (const float4*)pbso,     2);
    copy2d_v4<<<1500, 256, 0, stream>>>((float4*)out2, (const float4*)so_cls,   (2 * CC) / 4);
    copy2d_v4<<<1200, 256, 0, stream>>>((float4*)out3, (const float4*)rel_feat, DREL / 4);

    tdm_copy_roi<<<NB * NRr, 32, 0, stream>>>(so_roi, out4);

    gather_aux<<<8520, 512, 0, stream>>>(pro, pbx, cls_log, auxrel, roi, ent,
                                         out0, out1, out2, out3, out4);
}